// SmoothAP_20718922235946
// MI455X (gfx1250) — compile-verified
//
#include <hip/hip_runtime.h>
#include <hip/hip_bf16.h>

// SmoothAP loss for MI455X (gfx1250), exact-f32 path.
//   B=512, NUM_ID=64, FEAT=512, GROUP=8.
// S = f_s @ f_t^T via V_WMMA_F32_16X16X4_F32 (full f32 precision, matches reference).
// Only the diagonal 8x8 blocks of the B^3 ranking tensor are consumed, so the
// ranking step collapses to 4096 anchor sums over 512 columns.

typedef __attribute__((ext_vector_type(2))) float v2f;
typedef __attribute__((ext_vector_type(8))) float v8f;

#define BATCH 512
#define FEAT  512
#define GRP   8

// ---------------------------------------------------------------------------
// Kernel 1: S[512x512] = f_s @ f_t^T using f32 WMMA, one wave per 16x16 tile.
// A-layout (16x4 f32): lane l -> M = l%16, K = (l/16)*2 + {0,1}   (2 VGPRs)
// B-layout (4x16 f32): lane l -> N = l%16, K = (l/16)*2 + {0,1}   (2 VGPRs)
//   (B fed as rows of f_t, i.e. f_t^T columns)
// D-layout (16x16 f32, 8 VGPRs): lane l, vgpr v -> M = v + (l/16)*8, N = l%16
// ---------------------------------------------------------------------------
__global__ void __launch_bounds__(128)
smoothap_gemm_wmma(const float* __restrict__ fs,
                   const float* __restrict__ ft,
                   float* __restrict__ S) {
    const int wave = threadIdx.x >> 5;
    const int lane = threadIdx.x & 31;
    const int tile = blockIdx.x * 4 + wave;     // 0..1023
    const int trow = tile >> 5;                 // 0..31 (M tile)
    const int tcol = tile & 31;                 // 0..31 (N tile)

    const int mn   = lane & 15;                 // M (for A) / N (for B)
    const int koff = (lane >> 4) * 2;           // K offset within group of 4

    const float* aptr = fs + (size_t)(trow * 16 + mn) * FEAT + koff;
    const float* bptr = ft + (size_t)(tcol * 16 + mn) * FEAT + koff;

    v8f acc = {};
#pragma unroll 4
    for (int k = 0; k < FEAT; k += 4) {
        v2f a = *(const v2f*)(aptr + k);
        v2f b = *(const v2f*)(bptr + k);
        // (neg_a, A, neg_b, B, c_mod, C, reuse_a, reuse_b)
        acc = __builtin_amdgcn_wmma_f32_16x16x4_f32(
            false, a, false, b, (short)0, acc, false, false);
    }

    const int n     = lane & 15;
    const int mbase = (lane >> 4) * 8;
    float* out = S + (size_t)(trow * 16 + mbase) * BATCH + tcol * 16 + n;
#pragma unroll
    for (int v = 0; v < 8; ++v)
        out[(size_t)v * BATCH] = acc[v];
}

// ---------------------------------------------------------------------------
// Kernel 2: per-row ranking sums. One wave per row b.
//   rk_all(b,q) = 1 + sum_j relu(S[b,j] - S[b, base+q])     (mask term is 0)
//   rk_pos(b,q) = 1 + sum_{k<8} relu(S[b,base+k] - S[b,base+q])
//   partial[b]  = sum_q rk_pos/rk_all
// ---------------------------------------------------------------------------
__global__ void __launch_bounds__(256)
smoothap_rank(const float* __restrict__ S, float* __restrict__ partial) {
    const int wave = threadIdx.x >> 5;
    const int lane = threadIdx.x & 31;
    const int b = blockIdx.x * 8 + wave;        // 0..511

    const float* row = S + (size_t)b * BATCH;

    // Each lane caches 16 of the 512 columns (coalesced float4 loads).
    float vals[16];
#pragma unroll
    for (int c = 0; c < 4; ++c) {
        float4 v = ((const float4*)row)[c * 32 + lane];
        vals[c * 4 + 0] = v.x;
        vals[c * 4 + 1] = v.y;
        vals[c * 4 + 2] = v.z;
        vals[c * 4 + 3] = v.w;
    }

    // The 8 positive (same-identity) similarities — broadcast scalar loads.
    const int pbase = (b >> 3) * GRP;
    float pos[GRP];
#pragma unroll
    for (int k = 0; k < GRP; ++k) pos[k] = row[pbase + k];

    float acc = 0.0f;
#pragma unroll
    for (int q = 0; q < GRP; ++q) {
        const float a = pos[q];

        float s = 0.0f;
#pragma unroll
        for (int t = 0; t < 16; ++t)
            s += fmaxf(vals[t] - a, 0.0f);
        // wave32 reduction
#pragma unroll
        for (int off = 16; off > 0; off >>= 1)
            s += __shfl_xor(s, off, 32);
        const float rk_all = s + 1.0f;

        float rp = 0.0f;
#pragma unroll
        for (int k = 0; k < GRP; ++k)
            rp += fmaxf(pos[k] - a, 0.0f);

        acc += (rp + 1.0f) / rk_all;            // uniform across lanes
    }

    if (lane == 0) partial[b] = acc;
}

// ---------------------------------------------------------------------------
// Kernel 3: deterministic final reduction of 512 partials -> 1 - ap.
// ---------------------------------------------------------------------------
__global__ void __launch_bounds__(256)
smoothap_finalize(const float* __restrict__ partial, float* __restrict__ out) {
    __shared__ float sm[256];
    const int t = threadIdx.x;
    sm[t] = partial[t] + partial[t + 256];
    __syncthreads();
#pragma unroll
    for (int off = 128; off > 0; off >>= 1) {
        if (t < off) sm[t] += sm[t + off];
        __syncthreads();
    }
    if (t == 0) out[0] = 1.0f - sm[0] * (1.0f / 4096.0f);   // /GROUP/BATCH
}

extern "C" void kernel_launch(void* const* d_in, const int* in_sizes, int n_in,
                              void* d_out, int out_size, void* d_ws, size_t ws_size,
                              hipStream_t stream) {
    const float* f_s = (const float*)d_in[0];
    const float* f_t = (const float*)d_in[1];
    float* out = (float*)d_out;

    float* S       = (float*)d_ws;                    // 512*512 floats (1 MB)
    float* partial = S + (size_t)BATCH * BATCH;       // 512 floats

    // 1024 tiles (32x32), 4 waves (128 threads) per block.
    smoothap_gemm_wmma<<<256, 128, 0, stream>>>(f_s, f_t, S);
    // 512 rows, one wave each, 8 waves per block.
    smoothap_rank<<<64, 256, 0, stream>>>(S, partial);
    smoothap_finalize<<<1, 256, 0, stream>>>(partial, out);
}